// QLstm_18622978195480
// MI455X (gfx1250) — compile-verified
//
#include <hip/hip_runtime.h>
#include <math.h>

typedef float v2f __attribute__((ext_vector_type(2)));
typedef float v8f __attribute__((ext_vector_type(8)));

#define WIRES 4
#define NST   16      // 2^WIRES amplitudes
#define NQL   6       // six qlayers: f, i, Cg, o, h_new, y
#define TLEN  128
#define FEAT  16
#define DDIM  20      // FEAT + WIRES
#define BTOT  4096

__device__ __forceinline__ float sigm(float v) { return 1.f / (1.f + __expf(-v)); }

// ---------------------------------------------------------------------------
// Setup: build the six batch-invariant 16x16 complex unitaries U_k from phi.
// Thread t = (k, col): simulate basis column |col> through the gate sequence.
// ---------------------------------------------------------------------------
__global__ void build_unitaries(const float* __restrict__ phi,
                                float* __restrict__ uRe,
                                float* __restrict__ uIm) {
  int t = threadIdx.x;
  if (t >= NQL * NST) return;
  int k   = t >> 4;
  int col = t & 15;
  const float* w = phi + k * 24;   // (LAYERS=2, 12)

  float re[NST], im[NST];
  #pragma unroll
  for (int j = 0; j < NST; ++j) { re[j] = 0.f; im[j] = 0.f; }
  re[col] = 1.f;

  auto apply1 = [&](int q, float u00r, float u00i, float u01r, float u01i,
                            float u10r, float u10i, float u11r, float u11i) {
    int mbit = 1 << (3 - q);
    #pragma unroll
    for (int j = 0; j < NST; ++j) {
      if (j & mbit) continue;
      int j1 = j | mbit;
      float a0r = re[j],  a0i = im[j];
      float a1r = re[j1], a1i = im[j1];
      re[j]  = u00r*a0r - u00i*a0i + u01r*a1r - u01i*a1i;
      im[j]  = u00r*a0i + u00i*a0r + u01r*a1i + u01i*a1r;
      re[j1] = u10r*a0r - u10i*a0i + u11r*a1r - u11i*a1i;
      im[j1] = u10r*a0i + u10i*a0r + u11r*a1i + u11i*a1r;
    }
  };
  auto cnot = [&](int c, int tg) {
    int mc = 1 << (3 - c), mt = 1 << (3 - tg);
    #pragma unroll
    for (int j = 0; j < NST; ++j) {
      if ((j & mc) && !(j & mt)) {
        int j1 = j | mt;
        float r = re[j], i = im[j];
        re[j] = re[j1]; im[j] = im[j1];
        re[j1] = r;     im[j1] = i;
      }
    }
  };

  #pragma unroll
  for (int l = 0; l < 2; ++l) {
    #pragma unroll
    for (int i = 0; i < WIRES; ++i) cnot(i, (i + 1) & 3);
    #pragma unroll
    for (int i = 0; i < WIRES; ++i) cnot(i, (i + 2) & 3);
    #pragma unroll
    for (int i = 0; i < WIRES; ++i) {
      float tx = w[l*12 + 3*i + 0];
      float ty = w[l*12 + 3*i + 1];
      float tz = w[l*12 + 3*i + 2];
      float cx = cosf(0.5f*tx), sx = sinf(0.5f*tx);
      apply1(i, cx,0.f,  0.f,-sx,  0.f,-sx,  cx,0.f);   // RX
      float cy = cosf(0.5f*ty), sy = sinf(0.5f*ty);
      apply1(i, cy,0.f, -sy,0.f,   sy,0.f,   cy,0.f);   // RY
      float cz = cosf(0.5f*tz), sz = sinf(0.5f*tz);
      apply1(i, cz,-sz,  0.f,0.f,  0.f,0.f,  cz,sz);    // RZ
    }
  }
  #pragma unroll
  for (int j = 0; j < NST; ++j) {        // store U_k[row j][col]
    uRe[k*256 + j*16 + col] = re[j];
    uIm[k*256 + j*16 + col] = im[j];
  }
}

// ---------------------------------------------------------------------------
// Main kernel helpers
// ---------------------------------------------------------------------------
__device__ __forceinline__ v8f wmma4(v2f a, v2f b, v8f c) {
  // D(16x16,f32) = A(16x4,f32) * B(4x16,f32) + C
  return __builtin_amdgcn_wmma_f32_16x16x4_f32(
      false, a, false, b, (short)0, c, false, false);
}

// Complex 16x16 matvec batched over 16 rows, via f32 WMMA (full precision):
//   Re = Ar*Br + (-Ai)*Bi ; Im = Ar*Bi + Ai*Br
__device__ __forceinline__ void qlayer_mm(const v2f* aRe, const v2f* aIm,
                                          const v2f* aImN,
                                          const v2f* bR, const v2f* bI,
                                          v8f& cR, v8f& cI) {
  cR = v8f{0.f,0.f,0.f,0.f,0.f,0.f,0.f,0.f};
  cI = v8f{0.f,0.f,0.f,0.f,0.f,0.f,0.f,0.f};
  #pragma unroll
  for (int kk = 0; kk < 4; ++kk) cR = wmma4(aRe[kk],  bR[kk], cR);
  #pragma unroll
  for (int kk = 0; kk < 4; ++kk) cR = wmma4(aImN[kk], bI[kk], cR);
  #pragma unroll
  for (int kk = 0; kk < 4; ++kk) cI = wmma4(aRe[kk],  bI[kk], cI);
  #pragma unroll
  for (int kk = 0; kk < 4; ++kk) cI = wmma4(aIm[kk],  bR[kk], cI);
}

// C/D layout: VGPR r holds (M = r + 8*hi, N = lane&15); M=batch, N=state.
__device__ __forceinline__ void store_probs(const v8f& cR, const v8f& cI,
                                            float (*prob)[NST], int m, int hi) {
  #pragma unroll
  for (int r = 0; r < 8; ++r)
    prob[r + 8*hi][m] = cR[r]*cR[r] + cI[r]*cI[r];
}

// ev_q = sum_j (+/-)|amp_j|^2 ; wire-q bit of j is (j >> (3-q)) & 1
__device__ __forceinline__ void reduce_ev(float (*prob)[NST], int m, float* ev) {
  float e0 = 0.f, e1 = 0.f, e2 = 0.f, e3 = 0.f;
  #pragma unroll
  for (int j = 0; j < NST; ++j) {
    float p = prob[m][j];
    e0 = (j & 8) ? e0 - p : e0 + p;
    e1 = (j & 4) ? e1 - p : e1 + p;
    e2 = (j & 2) ? e2 - p : e2 + p;
    e3 = (j & 1) ? e3 - p : e3 + p;
  }
  ev[0] = e0; ev[1] = e1; ev[2] = e2; ev[3] = e3;
}

// Encoded product state: per wire, RZ(atan(a^2)) RY(atan(a)) |0>
__device__ __forceinline__ void encode_state(const float* a, float* pr, float* pi) {
  float vr[WIRES][2], vi[WIRES][2];
  #pragma unroll
  for (int q = 0; q < WIRES; ++q) {
    float t1 = atanf(a[q]);
    float t2 = atanf(a[q] * a[q]);
    float c1 = cosf(0.5f*t1), s1 = sinf(0.5f*t1);
    float c2 = cosf(0.5f*t2), s2 = sinf(0.5f*t2);
    vr[q][0] =  c1*c2;  vi[q][0] = -c1*s2;   // c1 * e^{-i t2/2}
    vr[q][1] =  s1*c2;  vi[q][1] =  s1*s2;   // s1 * e^{+i t2/2}
  }
  float cr[NST], ci[NST];
  cr[0] = 1.f; ci[0] = 0.f;
  #pragma unroll
  for (int q = 0; q < WIRES; ++q) {
    int len = 1 << q;
    #pragma unroll
    for (int idx = (1 << q) - 1; idx >= 0; --idx) {
      float ar = cr[idx], ai = ci[idx];
      #pragma unroll
      for (int b = 0; b < 2; ++b) {
        cr[idx*2 + b] = ar*vr[q][b] - ai*vi[q][b];
        ci[idx*2 + b] = ar*vi[q][b] + ai*vr[q][b];
      }
    }
    (void)len;
  }
  #pragma unroll
  for (int j = 0; j < NST; ++j) { pr[j] = cr[j]; pi[j] = ci[j]; }
}

// ---------------------------------------------------------------------------
// Main kernel: one wave32 per block; lanes 0-15 own 16 batch elements;
// all 32 lanes cooperate on the six complex 16x16 matvecs per step via WMMA.
// ---------------------------------------------------------------------------
__global__ void __launch_bounds__(32)
qlstm_kernel(const float* __restrict__ x,
             const float* __restrict__ Wd,
             const float* __restrict__ bd,
             const float* __restrict__ uRe,
             const float* __restrict__ uIm,
             float* __restrict__ out) {
  __shared__ float wdL[DDIM][4];
  __shared__ float bdL[4];
  __shared__ float psiRe[16][NST];
  __shared__ float psiIm[16][NST];
  __shared__ float prob[16][NST];

  const int lane = threadIdx.x;      // 0..31
  const int m    = lane & 15;        // batch-in-wave / fragment column
  const int hi   = lane >> 4;        // lane-half select
  const int bg   = blockIdx.x * 16 + m;   // global batch element (lanes 0-15)

  // Stage Wd[:, 0:4] and bd[0:4] (only these columns are ever consumed).
  for (int idx = lane; idx < DDIM * 4; idx += 32) {
    int r = idx >> 2, g = idx & 3;
    wdL[r][g] = Wd[r * DDIM + g];
  }
  if (lane < 4) bdL[lane] = bd[lane];
  __syncthreads();

  // B fragments = U_k^T slices, resident in VGPRs for the whole scan.
  // Assumed B(4x16) layout mirrors A(16x4): element (K = 4*kk + v + 2*hi, N = m).
  v2f bRe[NQL][4], bIm[NQL][4];
  #pragma unroll
  for (int k = 0; k < NQL; ++k)
    #pragma unroll
    for (int kk = 0; kk < 4; ++kk) {
      int s0 = 4*kk + 2*hi;                       // Bt[s][m] = U[m][s]
      v2f r, i2;
      r.x  = uRe[k*256 + m*16 + s0];  r.y  = uRe[k*256 + m*16 + s0 + 1];
      i2.x = uIm[k*256 + m*16 + s0];  i2.y = uIm[k*256 + m*16 + s0 + 1];
      bRe[k][kk] = r; bIm[k][kk] = i2;
    }

  float hreg[WIRES] = {0.f, 0.f, 0.f, 0.f};
  float creg[WIRES] = {0.f, 0.f, 0.f, 0.f};

  #pragma unroll 1
  for (int t = 0; t < TLEN; ++t) {
    // ---- dense 20x4 + tanh encoding + product state (lanes 0-15) ----
    if (hi == 0) {
      const float* xr = x + ((size_t)bg * TLEN + t) * FEAT;
      float z[4];
      #pragma unroll
      for (int g = 0; g < 4; ++g) z[g] = bdL[g];
      #pragma unroll
      for (int f = 0; f < FEAT; ++f) {
        float xv = xr[f];
        #pragma unroll
        for (int g = 0; g < 4; ++g) z[g] += xv * wdL[f][g];
      }
      #pragma unroll
      for (int q = 0; q < WIRES; ++q) {
        #pragma unroll
        for (int g = 0; g < 4; ++g) z[g] += hreg[q] * wdL[FEAT + q][g];
      }
      float Xf[4];
      #pragma unroll
      for (int g = 0; g < 4; ++g) Xf[g] = tanhf(0.5f * z[g]);  // 2*sigmoid(z)-1
      encode_state(Xf, psiRe[m], psiIm[m]);
    }
    __syncthreads();

    // A fragments: A[M=m][K = 4*kk + v + 2*hi] (documented 32-bit 16x4 layout)
    v2f aRe[4], aIm[4], aImN[4];
    #pragma unroll
    for (int kk = 0; kk < 4; ++kk) {
      int s0 = 4*kk + 2*hi;
      v2f r, i2, n;
      r.x  = psiRe[m][s0];  r.y  = psiRe[m][s0 + 1];
      i2.x = psiIm[m][s0];  i2.y = psiIm[m][s0 + 1];
      n.x = -i2.x; n.y = -i2.y;
      aRe[kk] = r; aIm[kk] = i2; aImN[kk] = n;
    }

    float evs[NQL][WIRES];
    #pragma unroll
    for (int k = 0; k < 4; ++k) {            // gates f, i, Cg, o
      v8f cR, cI;
      qlayer_mm(aRe, aIm, aImN, bRe[k], bIm[k], cR, cI);
      store_probs(cR, cI, prob, m, hi);
      __syncthreads();
      if (hi == 0) reduce_ev(prob, m, evs[k]);
      __syncthreads();
    }

    // ---- LSTM cell update + re-encode resc (lanes 0-15) ----
    if (hi == 0) {
      float resc[WIRES];
      #pragma unroll
      for (int q = 0; q < WIRES; ++q) {
        float fg = sigm(evs[0][q]);
        float ig = sigm(evs[1][q]);
        float cg = sigm(evs[2][q]);
        float og = sigm(evs[3][q]);
        creg[q] = fg * creg[q] + ig * cg;
        resc[q] = og * tanhf(creg[q]);
      }
      encode_state(resc, psiRe[m], psiIm[m]);
    }
    __syncthreads();

    #pragma unroll
    for (int kk = 0; kk < 4; ++kk) {
      int s0 = 4*kk + 2*hi;
      v2f r, i2, n;
      r.x  = psiRe[m][s0];  r.y  = psiRe[m][s0 + 1];
      i2.x = psiIm[m][s0];  i2.y = psiIm[m][s0 + 1];
      n.x = -i2.x; n.y = -i2.y;
      aRe[kk] = r; aIm[kk] = i2; aImN[kk] = n;
    }

    #pragma unroll
    for (int k = 4; k < 6; ++k) {            // h_new and y (no sigmoid)
      v8f cR, cI;
      qlayer_mm(aRe, aIm, aImN, bRe[k], bIm[k], cR, cI);
      store_probs(cR, cI, prob, m, hi);
      __syncthreads();
      if (hi == 0) reduce_ev(prob, m, evs[k]);
      __syncthreads();
    }

    if (hi == 0) {
      float* yout = out + ((size_t)bg * TLEN + t) * WIRES;
      #pragma unroll
      for (int q = 0; q < WIRES; ++q) {
        hreg[q] = evs[4][q];
        yout[q] = evs[5][q];
      }
    }
    __syncthreads();
  }

  // Final carries: out = [ys (B,T,4)] ++ [c (B,4)] ++ [h (B,4)]
  if (hi == 0) {
    const size_t ysN = (size_t)BTOT * TLEN * WIRES;
    #pragma unroll
    for (int q = 0; q < WIRES; ++q) {
      out[ysN + (size_t)bg * WIRES + q]                 = creg[q];
      out[ysN + (size_t)BTOT * WIRES + (size_t)bg * WIRES + q] = hreg[q];
    }
  }
}

// ---------------------------------------------------------------------------
extern "C" void kernel_launch(void* const* d_in, const int* in_sizes, int n_in,
                              void* d_out, int out_size, void* d_ws, size_t ws_size,
                              hipStream_t stream) {
  (void)in_sizes; (void)n_in; (void)out_size; (void)ws_size;
  const float* x   = (const float*)d_in[0];
  const float* phi = (const float*)d_in[1];
  const float* Wd  = (const float*)d_in[2];
  const float* bd  = (const float*)d_in[3];

  float* uRe = (float*)d_ws;            // 6 * 16 * 16 floats
  float* uIm = uRe + NQL * 256;         // 6 * 16 * 16 floats

  build_unitaries<<<1, NQL * NST, 0, stream>>>(phi, uRe, uIm);
  qlstm_kernel<<<BTOT / 16, 32, 0, stream>>>(x, Wd, bd, uRe, uIm, (float*)d_out);
}